// ProximityEffectCorrector_65111704207641
// MI455X (gfx1250) — compile-verified
//
#include <hip/hip_runtime.h>
#include <math.h>

// ---------------------------------------------------------------------------
// Butterworth HP sosfilt, [B=64, T=480000], 2 second-order sections.
// Exact parallel IIR via chunked superposition:
//   pass1: zero-state response per 480-sample chunk (+ 4-float state summary)
//   scan : s_{j+1} = A^L s_j + w_j  (64 parallel chains of 1000 steps)
//   pass3: y += C A^t s_j   == GEMM with K=4  -> V_WMMA_F32_16X16X4_F32
// ---------------------------------------------------------------------------

#define BATCH 64
#define TLEN  480000
#define LC    480                 // chunk length (16 | LC, LC*4 bytes 32B-aligned)
#define CPB   (TLEN / LC)         // 1000 chunks per batch row
#define NCH   (BATCH * CPB)       // 64000 chunks; batch stride = CPB*LC exactly
#define ROWT  (LC / 16)           // 30 row tiles per chunk
#define CTIL  (NCH / 16)          // 4000 column tiles

typedef __attribute__((ext_vector_type(2))) float v2f;
typedef __attribute__((ext_vector_type(8))) float v8f;

// ws layout (floats)
#define WS_M   0                  // 16    : A^LC row-major
#define WS_CA  16                 // 480*4 : rows C*A^t
#define WS_W   1936               // 64000*4 : zero-state chunk summaries
#define WS_S   257936             // 4*64000 : chunk-start states, S[k*NCH + c]

// --------------------------- precompute (1 thread) -------------------------
__global__ void hp4_precompute(const float* __restrict__ sos,
                               float* __restrict__ ws) {
  if (threadIdx.x != 0 || blockIdx.x != 0) return;
  const float b10 = sos[0],  b11 = sos[1],  b12 = sos[2];
  const float a11 = sos[4],  a12 = sos[5];
  const float b20 = sos[6],  b21 = sos[7],  b22 = sos[8];
  const float a21 = sos[10], a22 = sos[11];

  // state s = [u0,u1,v0,v1], s' = A s + B x, y = C s + D x
  float A[4][4] = {
      {-a11,            1.f, 0.f,  0.f},
      {-a12,            0.f, 0.f,  0.f},
      {b21 - a21 * b20, 0.f, -a21, 1.f},
      {b22 - a22 * b20, 0.f, -a22, 0.f}};

  // CA[t][k] = (C * A^t)[k],  C = [b20, 0, 1, 0]
  float r[4] = {b20, 0.f, 1.f, 0.f};
  float* CA = ws + WS_CA;
  for (int t = 0; t < LC; ++t) {
    CA[t * 4 + 0] = r[0]; CA[t * 4 + 1] = r[1];
    CA[t * 4 + 2] = r[2]; CA[t * 4 + 3] = r[3];
    float nr[4];
    for (int j = 0; j < 4; ++j)
      nr[j] = r[0] * A[0][j] + r[1] * A[1][j] + r[2] * A[2][j] + r[3] * A[3][j];
    r[0] = nr[0]; r[1] = nr[1]; r[2] = nr[2]; r[3] = nr[3];
  }

  // M = A^LC
  float M[4][4] = {{1, 0, 0, 0}, {0, 1, 0, 0}, {0, 0, 1, 0}, {0, 0, 0, 1}};
  for (int t = 0; t < LC; ++t) {
    float N[4][4];
    for (int i = 0; i < 4; ++i)
      for (int j = 0; j < 4; ++j)
        N[i][j] = M[i][0] * A[0][j] + M[i][1] * A[1][j] +
                  M[i][2] * A[2][j] + M[i][3] * A[3][j];
    for (int i = 0; i < 4; ++i)
      for (int j = 0; j < 4; ++j) M[i][j] = N[i][j];
  }
  float* Mo = ws + WS_M;
  for (int i = 0; i < 4; ++i)
    for (int j = 0; j < 4; ++j) Mo[i * 4 + j] = M[i][j];
}

// --------------------- pass 1: zero-state response per chunk ---------------
__global__ void hp4_pass1_zerostate(const float* __restrict__ x,
                                    const float* __restrict__ sos,
                                    float* __restrict__ yprov,
                                    float* __restrict__ ws) {
  const int c = blockIdx.x * blockDim.x + threadIdx.x;
  if (c >= NCH) return;
  const float b10 = sos[0],  b11 = sos[1],  b12 = sos[2];
  const float a11 = sos[4],  a12 = sos[5];
  const float b20 = sos[6],  b21 = sos[7],  b22 = sos[8];
  const float a21 = sos[10], a22 = sos[11];

  const float4* __restrict__ xp = (const float4*)(x + (size_t)c * LC);
  float4* __restrict__ yp = (float4*)(yprov + (size_t)c * LC);

  float u0 = 0.f, u1 = 0.f, v0 = 0.f, v1 = 0.f;
  auto step = [&](float xn) -> float {
    float y1  = fmaf(b10, xn, u0);
    float nu0 = fmaf(b11, xn, fmaf(-a11, y1, u1));
    float nu1 = fmaf(b12, xn, -a12 * y1);
    float y2  = fmaf(b20, y1, v0);
    float nv0 = fmaf(b21, y1, fmaf(-a21, y2, v1));
    float nv1 = fmaf(b22, y1, -a22 * y2);
    u0 = nu0; u1 = nu1; v0 = nv0; v1 = nv1;
    return y2;
  };

  for (int t = 0; t < LC / 4; ++t) {
    float4 xv = xp[t];
    float4 yv;
    yv.x = step(xv.x);
    yv.y = step(xv.y);
    yv.z = step(xv.z);
    yv.w = step(xv.w);
    yp[t] = yv;
  }
  float4 w = {u0, u1, v0, v1};
  ((float4*)(ws + WS_W))[c] = w;
}

// ------------------- pass 2: inter-chunk scan, 1 thread / batch ------------
__global__ void hp4_scan(float* __restrict__ ws) {
  const int b = threadIdx.x;
  if (b >= BATCH) return;
  const float* __restrict__ M = ws + WS_M;
  const float* __restrict__ W = ws + WS_W;
  float* __restrict__ S = ws + WS_S;
  float m[16];
#pragma unroll
  for (int i = 0; i < 16; ++i) m[i] = M[i];

  float s0 = 0.f, s1 = 0.f, s2 = 0.f, s3 = 0.f;
  int c = b * CPB;
  for (int j = 0; j < CPB; ++j, ++c) {
    S[0 * NCH + c] = s0;
    S[1 * NCH + c] = s1;
    S[2 * NCH + c] = s2;
    S[3 * NCH + c] = s3;
    const float4 w = ((const float4*)(W))[c];
    float t0 = fmaf(m[0],  s0, fmaf(m[1],  s1, fmaf(m[2],  s2, fmaf(m[3],  s3, w.x))));
    float t1 = fmaf(m[4],  s0, fmaf(m[5],  s1, fmaf(m[6],  s2, fmaf(m[7],  s3, w.y))));
    float t2 = fmaf(m[8],  s0, fmaf(m[9],  s1, fmaf(m[10], s2, fmaf(m[11], s3, w.z))));
    float t3 = fmaf(m[12], s0, fmaf(m[13], s1, fmaf(m[14], s2, fmaf(m[15], s3, w.w))));
    s0 = t0; s1 = t1; s2 = t2; s3 = t3;
  }
}

// -------- pass 3: y += C A^t s_j  as K=4 GEMM on V_WMMA_F32_16X16X4_F32 ----
__global__ void hp4_wmma_correct(float* __restrict__ y,
                                 const float* __restrict__ ws) {
  const int wave = (int)((blockIdx.x * blockDim.x + threadIdx.x) >> 5);
  const int lane = threadIdx.x & 31;
  const int trow = wave % ROWT;       // wave-uniform
  const int ccol = wave / ROWT;
  if (ccol >= CTIL) return;           // uniform per wave -> EXEC all-1s below
  const int t0 = trow * 16;
  const int j0 = ccol * 16;
  const int lh = lane & 15;
  const int hi = lane >> 4;

  const float* __restrict__ CA = ws + WS_CA;
  const float* __restrict__ S  = ws + WS_S;

  // A (16x4 f32): lanes 0-15 -> K=0,1 ; lanes 16-31 -> K=2,3 ; M = lane&15
  v2f a = *(const v2f*)(CA + 4 * (t0 + lh) + 2 * hi);

  // B (4x16 f32): vgpr0/1 hold rows (K, K+1), lane half selects K base 0 or 2
  const float* Srow = S + (size_t)(hi ? 2 : 0) * NCH + (j0 + lh);
  v2f b;
  b.x = Srow[0];
  b.y = Srow[NCH];

  // C/D (16x16 f32): lane holds column N=lane&15, rows M = v + 8*(lane>=16)
  float* base = y + (size_t)(j0 + lh) * LC + t0 + 8 * hi;   // 32B aligned
  v8f cacc = *(const v8f*)base;

  v8f d = __builtin_amdgcn_wmma_f32_16x16x4_f32(
      false, a, false, b, (short)0, cacc, false, false);

  *(v8f*)base = d;
}

// ---------------------------------------------------------------------------
extern "C" void kernel_launch(void* const* d_in, const int* in_sizes, int n_in,
                              void* d_out, int out_size, void* d_ws, size_t ws_size,
                              hipStream_t stream) {
  const float* audio = (const float*)d_in[0];   // [64, 480000]
  const float* sos   = (const float*)d_in[1];   // [2, 6]
  float* out = (float*)d_out;                   // [64, 480000]
  float* ws  = (float*)d_ws;                    // needs ~2.1 MB

  hp4_precompute<<<1, 32, 0, stream>>>(sos, ws);
  hp4_pass1_zerostate<<<NCH / 256, 256, 0, stream>>>(audio, sos, out, ws);
  hp4_scan<<<1, 64, 0, stream>>>(ws);
  // 30 * 4000 = 120000 tiles, 1 wave each, 8 waves per block
  hp4_wmma_correct<<<(ROWT * CTIL) / 8, 256, 0, stream>>>(out, ws);
}